// SingleHyperConv_49950469653065
// MI455X (gfx1250) — compile-verified
//
#include <hip/hip_runtime.h>
#include <hip/hip_bf16.h>

// ---------------- problem constants (match reference) ----------------
#define NN      65536       // nodes (== hyperedges M)
#define EE      1048576     // incidences
#define CC      128         // channels (IN_C == OUT_C)
#define NUM_G   64
#define NG      1024        // nodes per graph
#define KEEP    512         // K = NG/2 kept per graph
#define BN_EPS  1e-5f

typedef __attribute__((ext_vector_type(2))) float v2f;
typedef __attribute__((ext_vector_type(8))) float v8f;

// ====================================================================
// GEMM: out[N,128] = A[N,128] @ W[128,128] (+bias, optional leakyReLU)
// One wave32 per 16x16 tile via V_WMMA_F32_16X16X4_F32 (32 MMAs / tile).
// Block = 256 threads = 8 waves = one 16-row stripe across all 8 col tiles.
// ====================================================================
__global__ __launch_bounds__(256) void gemm_wmma_f32(
    const float* __restrict__ A, const float* __restrict__ W,
    const float* __restrict__ bias, int leaky, float* __restrict__ out)
{
    __shared__ float As[16 * CC];              // 8 KB A tile
    const int tid = threadIdx.x;
    const int r0  = blockIdx.x * 16;

    // stage A tile (16 rows x 128 cols) coalesced as float4
    const float4* A4  = reinterpret_cast<const float4*>(A);
    float4*       As4 = reinterpret_cast<float4*>(As);
    for (int p = tid; p < 16 * (CC / 4); p += 256) {
        int row = p >> 5, c4 = p & 31;         // 32 float4 per row
        As4[p] = A4[(size_t)(r0 + row) * (CC / 4) + c4];
    }
    __syncthreads();

    const int wave = tid >> 5;
    const int lane = tid & 31;
    const int half = lane >> 4;                // 0: lanes 0-15, 1: lanes 16-31
    const int l    = lane & 15;
    const int c0   = wave * 16;                // this wave's column tile

    v8f acc = {};
    for (int kk = 0; kk < CC; kk += 4) {
        const int ka = kk + 2 * half;          // VGPR0 holds K=0/K=2 halves
        v2f a, b;
        a.x = As[l * CC + ka];
        a.y = As[l * CC + ka + 1];
        b.x = W[(size_t)ka * CC + c0 + l];
        b.y = W[(size_t)(ka + 1) * CC + c0 + l];
        acc = __builtin_amdgcn_wmma_f32_16x16x4_f32(
                  false, a, false, b, (short)0, acc, false, false);
    }

    const float bv = bias ? bias[c0 + l] : 0.0f;
#pragma unroll
    for (int j = 0; j < 8; ++j) {              // VGPR j -> rows j and j+8
        int row = r0 + j + 8 * half;
        float v = acc[j] + bv;
        if (leaky) v = v > 0.0f ? v : 0.01f * v;
        out[(size_t)row * CC + c0 + l] = v;
    }
}

// ====================================================================
// degree accumulation: D[src] += hw[he];  Bdeg[he] += 1
// ====================================================================
__global__ void deg_kernel(const int* __restrict__ src, const int* __restrict__ he,
                           const float* __restrict__ hw,
                           float* __restrict__ Dacc, float* __restrict__ Bacc)
{
    int e = blockIdx.x * blockDim.x + threadIdx.x;
    if (e >= EE) return;
    atomicAdd(&Dacc[src[e]], hw[he[e]]);
    atomicAdd(&Bacc[he[e]], 1.0f);
}

__global__ void inv_kernel(const float* __restrict__ Dacc, const float* __restrict__ Bacc,
                           float* __restrict__ Dinv, float* __restrict__ Binv)
{
    int n = blockIdx.x * blockDim.x + threadIdx.x;
    if (n >= NN) return;
    float d = Dacc[n], b = Bacc[n];
    Dinv[n] = d > 0.0f ? 1.0f / d : 0.0f;
    Binv[n] = b > 0.0f ? 1.0f / b : 0.0f;
}

// ====================================================================
// generic feature scatter: acc[sidx[e], :] += feat[gidx[e], :] * scale[gidx[e]]
// 4 threads per edge, 32 channels each.
// ====================================================================
__global__ void scatter_add(const int* __restrict__ gidx, const int* __restrict__ sidx,
                            const float* __restrict__ feat, const float* __restrict__ scale,
                            float* __restrict__ acc)
{
    long long t = (long long)blockIdx.x * blockDim.x + threadIdx.x;
    int e = (int)(t >> 2);
    if (e >= EE) return;
    int cbase = ((int)t & 3) * 32;
    int g = gidx[e], s = sidx[e];
    float sc = scale ? scale[g] : 1.0f;
    const float* fp = feat + (size_t)g * CC + cbase;
    float*       ap = acc  + (size_t)s * CC + cbase;
#pragma unroll
    for (int c = 0; c < 32; ++c) atomicAdd(&ap[c], fp[c] * sc);
}

// ====================================================================
// BatchNorm statistics over z = Dinv[n]*zacc[n,c] + b_conv[c]
// ====================================================================
__global__ __launch_bounds__(256) void bn_stats(
    const float* __restrict__ zacc, const float* __restrict__ Dinv,
    const float* __restrict__ bconv, float* __restrict__ stats)
{
    __shared__ float ssum[256], ssq[256];
    int tid = threadIdx.x;
    int c = tid & (CC - 1), grp = tid >> 7;    // 2 row groups of 128 channels
    int r0 = blockIdx.x * 64;
    float s = 0.0f, q = 0.0f, bc = bconv[c];
    for (int r = grp; r < 64; r += 2) {
        int n = r0 + r;
        float v = Dinv[n] * zacc[(size_t)n * CC + c] + bc;
        s += v; q += v * v;
    }
    ssum[tid] = s; ssq[tid] = q;
    __syncthreads();
    if (grp == 0) {
        atomicAdd(&stats[c],      ssum[c] + ssum[c + CC]);
        atomicAdd(&stats[CC + c], ssq[c]  + ssq[c + CC]);
    }
}

__global__ void bn_finalize(float* __restrict__ stats, const float* __restrict__ gamma)
{
    int c = threadIdx.x;
    if (c >= CC) return;
    float mu  = stats[c] / (float)NN;
    float var = stats[CC + c] / (float)NN - mu * mu;
    stats[c]      = mu;
    stats[CC + c] = gamma[c] * rsqrtf(var + BN_EPS);   // fused scale
}

// z = leaky( (Dinv*zacc + b_conv - mu) * gamma*rstd + beta ) + res   (in place)
__global__ void bn_apply(float* __restrict__ z, const float* __restrict__ Dinv,
                         const float* __restrict__ bconv, const float* __restrict__ stats,
                         const float* __restrict__ beta, const float* __restrict__ res)
{
    size_t t = (size_t)blockIdx.x * blockDim.x + threadIdx.x;
    if (t >= (size_t)NN * CC) return;
    int c = (int)(t & (CC - 1));
    int n = (int)(t >> 7);
    float v = Dinv[n] * z[t] + bconv[c];
    v = (v - stats[c]) * stats[CC + c] + beta[c];
    v = v > 0.0f ? v : 0.01f * v;
    z[t] = v + res[t];
}

// ====================================================================
// SAGPooling score: tanh( dot(agg[n],W_rel) + b_rel + dot(z[n],W_root) )
// one wave32 per node, float4 per lane, shfl_xor reduction
// ====================================================================
__global__ __launch_bounds__(256) void score_kernel(
    const float* __restrict__ agg, const float* __restrict__ z,
    const float* __restrict__ Wrel, const float* __restrict__ brel,
    const float* __restrict__ Wroot, float* __restrict__ score)
{
    int tid = threadIdx.x, lane = tid & 31, wv = tid >> 5;
    int n = blockIdx.x * 8 + wv;
    if (n >= NN) return;
    const float4* a4 = reinterpret_cast<const float4*>(agg + (size_t)n * CC);
    const float4* z4 = reinterpret_cast<const float4*>(z   + (size_t)n * CC);
    const float4* wr = reinterpret_cast<const float4*>(Wrel);
    const float4* wo = reinterpret_cast<const float4*>(Wroot);
    float4 av = a4[lane], zv = z4[lane], rv = wr[lane], ov = wo[lane];
    float s = av.x * rv.x + av.y * rv.y + av.z * rv.z + av.w * rv.w
            + zv.x * ov.x + zv.y * ov.y + zv.z * ov.z + zv.w * ov.w;
#pragma unroll
    for (int m = 16; m; m >>= 1) s += __shfl_xor(s, m, 32);
    if (lane == 0) score[n] = tanhf(s + brel[0]);
}

// ====================================================================
// per-graph top-512 via block-wide bitonic sort of 1024 (score desc, idx asc)
// writes perm, mapping (pre-set -1) and pooled_batch (into d_out tail)
// ====================================================================
__global__ __launch_bounds__(1024) void topk_kernel(
    const float* __restrict__ score, const int* __restrict__ batch,
    int* __restrict__ perm, int* __restrict__ mapping, int* __restrict__ pooled_batch)
{
    __shared__ float sc[NG];
    __shared__ int   id[NG];
    int g = blockIdx.x, t = threadIdx.x;
    sc[t] = score[g * NG + t];
    id[t] = t;
    __syncthreads();
    for (int k = 2; k <= NG; k <<= 1) {
        for (int j = k >> 1; j > 0; j >>= 1) {
            int i = t ^ j;
            if (i > t) {
                bool dirDesc = (t & k) == 0;
                float s0 = sc[t], s1 = sc[i];
                int   i0 = id[t], i1 = id[i];
                bool before = (s0 > s1) || (s0 == s1 && i0 < i1);   // top_k order
                if (before != dirDesc) { sc[t] = s1; sc[i] = s0; id[t] = i1; id[i] = i0; }
            }
            __syncthreads();
        }
    }
    if (t < KEEP) {
        int node = g * NG + id[t];
        int pi = g * KEEP + t;
        perm[pi]         = node;
        mapping[node]    = pi;
        pooled_batch[pi] = batch[node];
    }
}

__global__ void pool_kernel(const float* __restrict__ z, const float* __restrict__ score,
                            const int* __restrict__ perm, float* __restrict__ out)
{
    size_t t = (size_t)blockIdx.x * blockDim.x + threadIdx.x;
    if (t >= (size_t)NUM_G * KEEP * CC) return;
    int i = (int)(t >> 7), c = (int)(t & (CC - 1));
    int node = perm[i];
    out[t] = z[(size_t)node * CC + c] * score[node];
}

__global__ void edges_kernel(const int* __restrict__ src, const int* __restrict__ he,
                             const int* __restrict__ mapping, int* __restrict__ out)
{
    int e = blockIdx.x * blockDim.x + threadIdx.x;
    if (e >= EE) return;
    int ns = mapping[src[e]], nd = mapping[he[e]];
    bool valid = (ns >= 0) && (nd >= 0);
    out[e]      = valid ? ns : -1;
    out[EE + e] = valid ? nd : -1;
}

// ====================================================================
extern "C" void kernel_launch(void* const* d_in, const int* in_sizes, int n_in,
                              void* d_out, int out_size, void* d_ws, size_t ws_size,
                              hipStream_t stream)
{
    (void)in_sizes; (void)n_in; (void)out_size; (void)ws_size;

    const float* x      = (const float*)d_in[0];
    const int*   hei    = (const int*)  d_in[1];
    const float* hw     = (const float*)d_in[2];
    const int*   batch  = (const int*)  d_in[3];
    const float* Wconv  = (const float*)d_in[4];
    const float* bconv  = (const float*)d_in[5];
    const float* gamma  = (const float*)d_in[6];
    const float* beta   = (const float*)d_in[7];
    const float* Wlin1  = (const float*)d_in[8];
    const float* blin1  = (const float*)d_in[9];
    const float* Wrel   = (const float*)d_in[10];
    const float* brel   = (const float*)d_in[11];
    const float* Wroot  = (const float*)d_in[12];

    const int* src = hei;        // hyperedge_index[0]
    const int* he  = hei + EE;   // hyperedge_index[1]

    // ---- workspace layout (bytes) ----
    char* ws = (char*)d_ws;
    const size_t FEAT = (size_t)NN * CC * sizeof(float);   // 32 MB
    float* xt_agg = (float*)(ws);                          // xt, later reused as agg
    float* ef     = (float*)(ws + FEAT);                   // edge features
    float* res    = (float*)(ws + 2 * FEAT);               // residual branch
    float* z      = (float*)(ws + 3 * FEAT);               // z accumulator / final z
    char*  sc     = ws + 4 * FEAT;
    float* Dacc   = (float*)(sc);                          // N
    float* Bacc   = (float*)(sc + 1 * NN * 4);             // N  (contiguous with Dacc)
    float* Dinv   = (float*)(sc + 2 * NN * 4);
    float* Binv   = (float*)(sc + 3 * NN * 4);
    float* score  = (float*)(sc + 4 * NN * 4);
    float* stats  = (float*)(sc + 5 * NN * 4);             // 256 floats
    int*   perm   = (int*)  (sc + 5 * NN * 4 + 1024);      // 32768 ints
    int*   mapping= (int*)  (sc + 5 * NN * 4 + 1024 + NUM_G * KEEP * 4);

    // ---- output layout: [z_pool 32768*128 f32][edge_index 2*E i32][batch 32768 i32]
    float* out_zpool = (float*)d_out;
    int*   out_edges = (int*)d_out + (size_t)NUM_G * KEEP * CC;
    int*   out_batch = out_edges + 2 * (size_t)EE;

    // ---- zero / init scratch (capture-legal async memsets) ----
    hipMemsetAsync(Dacc, 0, 2 * (size_t)NN * 4, stream);       // Dacc + Bacc
    hipMemsetAsync(ef,   0, FEAT, stream);
    hipMemsetAsync(z,    0, FEAT, stream);
    hipMemsetAsync(stats, 0, 256 * 4, stream);
    hipMemsetAsync(mapping, 0xFF, (size_t)NN * 4, stream);     // -1

    // ---- 1) dense GEMMs (WMMA): xt = x@Wconv ; res = leaky(x@Wlin1 + blin1)
    gemm_wmma_f32<<<NN / 16, 256, 0, stream>>>(x, Wconv, nullptr, 0, xt_agg);
    gemm_wmma_f32<<<NN / 16, 256, 0, stream>>>(x, Wlin1, blin1, 1, res);

    // ---- 2) degrees & inverses
    deg_kernel<<<EE / 256, 256, 0, stream>>>(src, he, hw, Dacc, Bacc);
    inv_kernel<<<NN / 256, 256, 0, stream>>>(Dacc, Bacc, Dinv, Binv);

    // ---- 3) edge_feat[he] += xt[src]   (B^-1 applied at read time in next pass)
    scatter_add<<<(4 * EE) / 256, 256, 0, stream>>>(src, he, xt_agg, nullptr, ef);

    // ---- 4) z[src] += Binv[he] * edge_feat[he]
    scatter_add<<<(4 * EE) / 256, 256, 0, stream>>>(he, src, ef, Binv, z);

    // ---- 5) BatchNorm (train-mode stats) + leaky + residual, in place on z
    bn_stats<<<NN / 64, 256, 0, stream>>>(z, Dinv, bconv, stats);
    bn_finalize<<<1, CC, 0, stream>>>(stats, gamma);
    bn_apply<<<(NN * CC) / 256, 256, 0, stream>>>(z, Dinv, bconv, stats, beta, res);

    // ---- 6) scorer aggregation: agg[he] += z[src]  (reuse xt buffer)
    hipMemsetAsync(xt_agg, 0, FEAT, stream);
    scatter_add<<<(4 * EE) / 256, 256, 0, stream>>>(src, he, z, nullptr, xt_agg);

    // ---- 7) score = tanh(agg@W_rel + b_rel + z@W_root)
    score_kernel<<<NN / 8, 256, 0, stream>>>(xt_agg, z, Wrel, brel, Wroot, score);

    // ---- 8) per-graph top-K (bitonic), perm/mapping/pooled_batch
    topk_kernel<<<NUM_G, NG, 0, stream>>>(score, batch, perm, mapping, out_batch);

    // ---- 9) z_pool = z[perm] * score[perm] ; edge remap
    pool_kernel<<<((size_t)NUM_G * KEEP * CC) / 256, 256, 0, stream>>>(z, score, perm, out_zpool);
    edges_kernel<<<EE / 256, 256, 0, stream>>>(src, he, mapping, out_edges);
}